// ResnetHyper_31026843747004
// MI455X (gfx1250) — compile-verified
//
#include <hip/hip_runtime.h>
#include <hip/hip_bf16.h>
#include <stddef.h>

typedef __attribute__((ext_vector_type(2))) float v2f;
typedef __attribute__((ext_vector_type(8))) float v8f;

#define HID       512
#define C_REAL    105
#define C_PAD     112            // 7 tiles of 16
#define C_TILES   7
#define KW_TOTAL  220000         // 11 * 20000
#define KW_TILES  (KW_TOTAL / 16) // 13750, exact
#define WAVES_PER_BLOCK 8

// ---------------- MLP layer 1: x = [pref_embedding | chunk_emb[c]] -> relu(W1 x + b1) ----
__global__ __launch_bounds__(256)
void mlp_l1(const float* __restrict__ pref, const float* __restrict__ pref_emb,
            const float* __restrict__ chunk_emb, const float* __restrict__ W1,
            const float* __restrict__ b1, float* __restrict__ h1) {
    int g = blockIdx.x * blockDim.x + threadIdx.x;
    if (g >= C_PAD * HID) return;
    int c = g % C_PAD;
    int j = g / C_PAD;
    float s = 0.0f;
    if (c < C_REAL) {
        float p0 = pref[0], p1 = pref[1];
        const float* w = W1 + j * 128;           // W1 row j: [128]
        const float* ce = chunk_emb + c * 64;
        #pragma unroll 4
        for (int i = 0; i < 64; ++i) {
            float pe = fmaf(p1, pref_emb[64 + i], p0 * pref_emb[i]);
            s = fmaf(pe, w[i], s);
            s = fmaf(ce[i], w[64 + i], s);
        }
        s = fmaxf(s + b1[j], 0.0f);
    }
    h1[c * HID + j] = s;   // zero-pad rows 105..111
}

// ---------------- MLP layers 2/3: hout = (relu?)(hin @ W^T + b) --------------------------
template <bool RELU>
__global__ __launch_bounds__(256)
void mlp_layer(const float* __restrict__ hin, const float* __restrict__ W,
               const float* __restrict__ b, float* __restrict__ hout) {
    int g = blockIdx.x * blockDim.x + threadIdx.x;
    if (g >= C_PAD * HID) return;
    int c = g % C_PAD;
    int j = g / C_PAD;
    float s = 0.0f;
    if (c < C_REAL) {
        const float* w = W + j * HID;
        const float* x = hin + c * HID;
        #pragma unroll 8
        for (int i = 0; i < HID; ++i) s = fmaf(x[i], w[i], s);
        s += b[j];
        if (RELU) s = fmaxf(s, 0.0f);
    }
    hout[c * HID + j] = s;
}

// ---------------- Big memory-bound GEMM: out[105,220000] = rep @ ws_flat^T ---------------
// One wave owns a 16-wide kw tile of ws (B) and computes all 7 c-tiles (A) against it,
// so ws (450 MB) is streamed exactly once from HBM. fp32 WMMA 16x16x4 keeps reference
// precision. K loop: 512/4 = 128 steps, 7 accumulators.
__global__ __launch_bounds__(256)
void gemm_rep_ws(const float* __restrict__ rep,   // [112][512], rows >=105 zero
                 const float* __restrict__ ws,    // [220000][512]
                 float* __restrict__ out) {       // [105][220000]
    const int wave = threadIdx.x >> 5;   // wave32
    const int lane = threadIdx.x & 31;
    const int tile = blockIdx.x * WAVES_PER_BLOCK + wave;
    if (tile >= KW_TILES) return;        // wave-uniform: EXEC stays all-ones below

    const int kw0  = tile * 16;
    const int half = lane >> 4;          // 0: K pair {0,1}; 1: K pair {2,3}
    const int l16  = lane & 15;

    // B fragment source: lane -> column N=l16 of the kw tile, consecutive h pair
    const float* bptr = ws + (size_t)(kw0 + l16) * HID + (half << 1);
    // A fragment source: lane -> row M=l16 of a c-tile, consecutive h pair
    const float* aptr = rep + (size_t)l16 * HID + (half << 1);

    v8f acc[C_TILES];
    #pragma unroll
    for (int ct = 0; ct < C_TILES; ++ct) acc[ct] = (v8f){};

    #pragma unroll 2
    for (int k = 0; k < HID; k += 4) {
        const v2f b = *(const v2f*)(bptr + k);                 // streamed once from HBM
        #pragma unroll
        for (int ct = 0; ct < C_TILES; ++ct) {
            const v2f a = *(const v2f*)(aptr + (size_t)ct * 16 * HID + k); // L0/L2 hot
            acc[ct] = __builtin_amdgcn_wmma_f32_16x16x4_f32(
                false, a, false, b, (short)0, acc[ct], false, false);
        }
    }

    // D layout: VGPR i -> M = i + 8*half (lanes 0-15: M=i, lanes 16-31: M=i+8), N = l16
    const size_t col = (size_t)kw0 + l16;
    #pragma unroll
    for (int ct = 0; ct < C_TILES; ++ct) {
        const int cbase = ct * 16 + half * 8;
        #pragma unroll
        for (int i = 0; i < 8; ++i) {
            const int c = cbase + i;
            if (c < C_REAL) out[(size_t)c * KW_TOTAL + col] = acc[ct][i];
        }
    }
}

extern "C" void kernel_launch(void* const* d_in, const int* in_sizes, int n_in,
                              void* d_out, int out_size, void* d_ws, size_t ws_size,
                              hipStream_t stream) {
    const float* pref      = (const float*)d_in[0];
    const float* pref_emb  = (const float*)d_in[1];
    const float* chunk_emb = (const float*)d_in[2];
    const float* W1        = (const float*)d_in[3];
    const float* b1        = (const float*)d_in[4];
    const float* W2        = (const float*)d_in[5];
    const float* b2        = (const float*)d_in[6];
    const float* W3        = (const float*)d_in[7];
    const float* b3        = (const float*)d_in[8];
    const float* ws        = (const float*)d_in[9];
    float* out = (float*)d_out;

    float* h1  = (float*)d_ws;                 // [112][512]
    float* h2  = h1 + C_PAD * HID;             // [112][512]
    float* rep = h2 + C_PAD * HID;             // [112][512]

    const int n = C_PAD * HID;                 // 57344
    const int mlp_blocks = (n + 255) / 256;    // 224

    mlp_l1<<<mlp_blocks, 256, 0, stream>>>(pref, pref_emb, chunk_emb, W1, b1, h1);
    mlp_layer<true ><<<mlp_blocks, 256, 0, stream>>>(h1, W2, b2, h2);
    mlp_layer<false><<<mlp_blocks, 256, 0, stream>>>(h2, W3, b3, rep);

    const int gemm_blocks = (KW_TILES + WAVES_PER_BLOCK - 1) / WAVES_PER_BLOCK; // 1719
    gemm_rep_ws<<<gemm_blocks, 256, 0, stream>>>(rep, ws, out);
}